// PointCloud_SDF_86182813761644
// MI455X (gfx1250) — compile-verified
//
#include <hip/hip_runtime.h>

typedef float v2f __attribute__((ext_vector_type(2)));
typedef float v8f __attribute__((ext_vector_type(8)));

#define WAVES_PER_WG 8
#define THREADS (WAVES_PER_WG * 32)
#define CHUNK_PTS 4096   // 4096 pts * 16B = 64KB LDS
#define TILE 16
#define SPHERE_RADIUS 0.05f

static __device__ __forceinline__ unsigned umin32(unsigned a, unsigned b) {
    return a < b ? a : b;   // -> v_min_u32 / v_min3_u32, no canonicalize
}
static __device__ __forceinline__ unsigned f2u(float f) {
    union { float f; unsigned u; } c; c.f = f; return c.u;
}
static __device__ __forceinline__ float u2f(unsigned u) {
    union { float f; unsigned u; } c; c.u = u; return c.f;
}

// 1-NN squared distance via WMMA f32 16x16x4:
//   A row m   = (x0, x1, x2, 1.0)
//   B col n   = (-2p0, -2p1, -2p2, ||p||^2)
//   C[m][n]   = ||x||^2
//   D[m][n]   = ||x - p||^2  (>= 0)  -> running min as u32 (order-isomorphic)
__global__ __launch_bounds__(THREADS)
void pcd_sdf_wmma_kernel(const float* __restrict__ x,
                         const float* __restrict__ pcd,
                         const float* __restrict__ signs,
                         float* __restrict__ out,
                         int M, int N) {
    __shared__ float lds[CHUNK_PTS * 4];

    const int tid  = threadIdx.x;
    const int wave = tid >> 5;
    const int lane = tid & 31;
    const int half = lane >> 4;   // 0: lanes 0-15, 1: lanes 16-31
    const int sub  = lane & 15;

    const int rowBase = (blockIdx.x * WAVES_PER_WG + wave) * 16;

    // ---- A fragment (ISA 32-bit A 16x4 layout):
    // lane<16 : (A[m][K0], A[m][K1]) = (x0, x1),  m = sub
    // lane>=16: (A[m][K2], A[m][K3]) = (x2, 1.0)
    v2f a;
    {
        const float* xr = x + (size_t)(rowBase + sub) * 3;
        if (half == 0) { a.x = xr[0]; a.y = xr[1]; }
        else           { a.x = xr[2]; a.y = 1.0f; }
    }

    // ---- C fragment: C[m][n] = ||x_m||^2, built from per-lane partials.
    // lane<16 holds x0^2+x1^2 of row sub; lane>=16 holds x2^2 of row sub.
    const float part = (half == 0) ? (a.x * a.x + a.y * a.y) : (a.x * a.x);
    v8f cxsq;
#pragma unroll
    for (int i = 0; i < 8; ++i) {
        const int r = i + 8 * half;                 // row this lane holds in D
        cxsq[i] = __shfl(part, r, 32) + __shfl(part, r + 16, 32);
    }

    unsigned best[8];
#pragma unroll
    for (int i = 0; i < 8; ++i) best[i] = 0x7F800000u;   // +inf

    for (int cbase = 0; cbase < N; cbase += CHUNK_PTS) {
        __syncthreads();
        // preprocess chunk into packed B records in LDS
#pragma unroll
        for (int i = 0; i < CHUNK_PTS / THREADS; ++i) {
            const int p = tid + i * THREADS;
            const float* pp = pcd + (size_t)(cbase + p) * 3;
            const float p0 = pp[0], p1 = pp[1], p2 = pp[2];
            float* dst = lds + p * 4;
            dst[0] = -2.0f * p0;
            dst[1] = -2.0f * p1;
            dst[2] = -2.0f * p2;
            dst[3] = p0 * p0 + p1 * p1 + p2 * p2;
        }
        __syncthreads();

        // ---- B fragment (ISA 32-bit B 4x16 layout):
        // lane<16 : (B[K0][n], B[K1][n]) = (-2p0, -2p1),  n = sub
        // lane>=16: (B[K2][n], B[K3][n]) = (-2p2, psq)
#pragma unroll 2
        for (int t = 0; t < CHUNK_PTS / TILE; t += 2) {
            const float* bp = lds + (t * TILE + sub) * 4 + half * 2;
            v2f b0, b1;
            b0.x = bp[0];            b0.y = bp[1];
            b1.x = bp[TILE * 4 + 0]; b1.y = bp[TILE * 4 + 1];
            v8f d0 = __builtin_amdgcn_wmma_f32_16x16x4_f32(
                false, a, false, b0, (short)0, cxsq, false, false);
            v8f d1 = __builtin_amdgcn_wmma_f32_16x16x4_f32(
                false, a, false, b1, (short)0, cxsq, false, false);
#pragma unroll
            for (int i = 0; i < 8; ++i)
                best[i] = umin32(best[i], umin32(f2u(d0[i]), f2u(d1[i])));
        }
    }

    // cross-lane min inside each 16-lane half (rows 0-7 live in lanes 0-15,
    // rows 8-15 in lanes 16-31, per the 16x16 f32 C/D layout)
#pragma unroll
    for (int i = 0; i < 8; ++i) {
        unsigned v = best[i];
        v = umin32(v, (unsigned)__shfl_xor((int)v, 1, 32));
        v = umin32(v, (unsigned)__shfl_xor((int)v, 2, 32));
        v = umin32(v, (unsigned)__shfl_xor((int)v, 4, 32));
        v = umin32(v, (unsigned)__shfl_xor((int)v, 8, 32));
        best[i] = v;
    }

    // lane 0 writes rows 0-7 of this block, lane 16 writes rows 8-15
    if (sub == 0) {
#pragma unroll
        for (int i = 0; i < 8; ++i) {
            const int m = rowBase + half * 8 + i;
            float dmin = u2f(best[i]);
            dmin = dmin > 0.0f ? dmin : 0.0f;   // also kills -eps (huge u32 never wins)
            out[m] = (__builtin_sqrtf(dmin) - SPHERE_RADIUS) * signs[m];
        }
    }
}

extern "C" void kernel_launch(void* const* d_in, const int* in_sizes, int n_in,
                              void* d_out, int out_size, void* d_ws, size_t ws_size,
                              hipStream_t stream) {
    const float* x     = (const float*)d_in[0];   // (M,3)
    const float* pcd   = (const float*)d_in[1];   // (N,3)
    const float* signs = (const float*)d_in[2];   // (M,1)
    float* out = (float*)d_out;                   // (M,1)

    const int M = in_sizes[0] / 3;
    const int N = in_sizes[1] / 3;

    const int rowsPerWG = 16 * WAVES_PER_WG;      // 128
    const int grid = M / rowsPerWG;               // 512 for M=65536

    pcd_sdf_wmma_kernel<<<grid, THREADS, 0, stream>>>(x, pcd, signs, out, M, N);
}